// DistMatchLayer_v3_2_73461120631383
// MI455X (gfx1250) — compile-verified
//
#include <hip/hip_runtime.h>
#include <math.h>

typedef __attribute__((ext_vector_type(2))) float v2f;
typedef __attribute__((ext_vector_type(8))) float v8f;

#define BATCH 4
#define NA    8192
#define NB    8192
#define NCH   112
#define TOPK  5
#define CHUNK 2048          // candidates staged in LDS per block iteration (32 KB)
#define NTILE (CHUNK / 16)  // 16-candidate tiles per chunk
#define QPB   128           // queries per block = 8 waves * 16
#define INF_F 3.0e38f

__global__ __launch_bounds__(256)
void distmatch_topk_wmma(const int* __restrict__ coords_a,
                         const int* __restrict__ coords_b,
                         const float* __restrict__ feats_a,
                         const float* __restrict__ feats_b,
                         const float* __restrict__ w1,
                         const float* __restrict__ b1,
                         float* __restrict__ out)
{
    __shared__ float candTab[CHUNK * 4];      // (-2x,-2y,-2z,|c|^2) per candidate
    __shared__ float sTmpd[8][16][TOPK];
    __shared__ int   sIdx [8][16][TOPK];

    const int tid  = threadIdx.x;
    const int wave = tid >> 5;
    const int lane = tid & 31;
    const int col  = lane & 15;               // query column / A row
    const int half = lane >> 4;

    const int blocksPerBatch = NA / QPB;
    const int b  = blockIdx.x / blocksPerBatch;
    const int n0 = (blockIdx.x % blocksPerBatch) * QPB + wave * 16;
    const int q  = n0 + col;

    // Query coords (B operand, fixed for whole scan).
    const int* qc = coords_a + ((size_t)b * NA + q) * 3;
    const float qx = (float)qc[0];
    const float qy = (float)qc[1];
    const float qz = (float)qc[2];
    const float qnorm = qx*qx + qy*qy + qz*qz;

    // B 4x16 layout: VGPR0 = {K0 | K2}, VGPR1 = {K1 | K3} split by lane half.
    v2f bop;
    bop[0] = half ? qz : qx;
    bop[1] = half ? 1.0f : qy;

    // Per-lane sorted ascending top-5 of s = |c|^2 - 2 c.q  (same order as d2).
    float tv0=INF_F, tv1=INF_F, tv2=INF_F, tv3=INF_F, tv4=INF_F;
    int   ti0=0x7fffffff, ti1=0x7fffffff, ti2=0x7fffffff, ti3=0x7fffffff, ti4=0x7fffffff;

    // Branchless sorted insert of one candidate (s, ci) into the top-5.
    auto insert8 = [&](const v8f& d, int ibase) {
        #pragma unroll
        for (int j = 0; j < 8; ++j) {
            const float s  = d[j];
            const int   ci = ibase + j;
            const bool lt0 = s < tv0, lt1 = s < tv1, lt2 = s < tv2,
                       lt3 = s < tv3, lt4 = s < tv4;
            const float n4 = lt4 ? (lt3 ? tv3 : s ) : tv4;
            const int   m4 = lt4 ? (lt3 ? ti3 : ci) : ti4;
            const float n3 = lt3 ? (lt2 ? tv2 : s ) : tv3;
            const int   m3 = lt3 ? (lt2 ? ti2 : ci) : ti3;
            const float n2 = lt2 ? (lt1 ? tv1 : s ) : tv2;
            const int   m2 = lt2 ? (lt1 ? ti1 : ci) : ti2;
            const float n1 = lt1 ? (lt0 ? tv0 : s ) : tv1;
            const int   m1 = lt1 ? (lt0 ? ti0 : ci) : ti1;
            const float nz = lt0 ? s  : tv0;
            const int   mz = lt0 ? ci : ti0;
            tv0=nz; tv1=n1; tv2=n2; tv3=n3; tv4=n4;
            ti0=mz; ti1=m1; ti2=m2; ti3=m3; ti4=m4;
        }
    };

    for (int cbase = 0; cbase < NB; cbase += CHUNK) {
        // Cooperative chunk load: int coords -> prefolded float4 in LDS.
        for (int i = tid; i < CHUNK; i += 256) {
            const int* cc = coords_b + ((size_t)b * NB + (cbase + i)) * 3;
            const float x = (float)cc[0], y = (float)cc[1], z = (float)cc[2];
            candTab[i*4+0] = -2.0f * x;
            candTab[i*4+1] = -2.0f * y;
            candTab[i*4+2] = -2.0f * z;
            candTab[i*4+3] = x*x + y*y + z*z;
        }
        __syncthreads();

        // A 16x4 layout: lane carries row M=col; VGPRs = {K0,K1} (half 0) / {K2,K3} (half 1).
        const float* tabl = candTab + (size_t)col * 4 + half * 2;

        // Software-pipelined, 2 tiles (32 candidates) per iteration.
        v2f aCur0 = *(const v2f*)(tabl + 0);
        v2f aCur1 = *(const v2f*)(tabl + 64);
        for (int t = 0; t < NTILE; t += 2) {
            const int tn = (t + 2 < NTILE) ? (t + 2) : t;   // harmless re-read on last trip
            v2f aNxt0 = *(const v2f*)(tabl + (size_t)tn * 64);
            v2f aNxt1 = *(const v2f*)(tabl + (size_t)(tn + 1) * 64);

            v8f cz0 = {};
            v8f d0 = __builtin_amdgcn_wmma_f32_16x16x4_f32(
                false, aCur0, false, bop, (short)0, cz0, false, false);
            v8f cz1 = {};
            v8f d1 = __builtin_amdgcn_wmma_f32_16x16x4_f32(
                false, aCur1, false, bop, (short)0, cz1, false, false);

            const float m0 = fminf(fminf(fminf(d0[0],d0[1]), fminf(d0[2],d0[3])),
                                   fminf(fminf(d0[4],d0[5]), fminf(d0[6],d0[7])));
            const float m1 = fminf(fminf(fminf(d1[0],d1[1]), fminf(d1[2],d1[3])),
                                   fminf(fminf(d1[4],d1[5]), fminf(d1[6],d1[7])));
            if (__any(fminf(m0, m1) < tv4)) {
                insert8(d0, cbase + t*16 + half*8);          // D: VGPR j -> row j + 8*half
                insert8(d1, cbase + (t+1)*16 + half*8);
            }
            aCur0 = aNxt0;
            aCur1 = aNxt1;
        }
        __syncthreads();
    }

    // Merge the two half-lane lists for each query (both halves compute same result).
    float a0=tv0,a1=tv1,a2=tv2,a3=tv3,a4=tv4;
    int   x0=ti0,x1=ti1,x2=ti2,x3=ti3,x4=ti4;
    float c0v=__shfl_xor(tv0,16), c1v=__shfl_xor(tv1,16), c2v=__shfl_xor(tv2,16),
          c3v=__shfl_xor(tv3,16), c4v=__shfl_xor(tv4,16);
    int   y0=__shfl_xor(ti0,16), y1=__shfl_xor(ti1,16), y2=__shfl_xor(ti2,16),
          y3=__shfl_xor(ti3,16), y4=__shfl_xor(ti4,16);

    float rv[TOPK]; int ri[TOPK];
    #pragma unroll
    for (int k = 0; k < TOPK; ++k) {
        const bool ta = (a0 < c0v) || (a0 == c0v && x0 < y0);
        rv[k] = ta ? a0 : c0v;
        ri[k] = ta ? x0 : y0;
        const float pa0 = ta ? a1 : a0, pa1 = ta ? a2 : a1,
                    pa2 = ta ? a3 : a2, pa3 = ta ? a4 : a3;
        const int   px0 = ta ? x1 : x0, px1 = ta ? x2 : x1,
                    px2 = ta ? x3 : x2, px3 = ta ? x4 : x3;
        const float pb0 = ta ? c0v : c1v, pb1 = ta ? c1v : c2v,
                    pb2 = ta ? c2v : c3v, pb3 = ta ? c3v : c4v;
        const int   py0 = ta ? y0 : y1, py1 = ta ? y1 : y2,
                    py2 = ta ? y2 : y3, py3 = ta ? y3 : y4;
        a0=pa0; a1=pa1; a2=pa2; a3=pa3;  x0=px0; x1=px1; x2=px2; x3=px3;
        c0v=pb0; c1v=pb1; c2v=pb2; c3v=pb3;  y0=py0; y1=py1; y2=py2; y3=py3;
    }

    // Finalize: d2 = s + |q|^2 ; tmp_d = (0.5 - clip(sqrt(d2)/32, 0, 0.5)) * 2
    #pragma unroll
    for (int k = 0; k < TOPK; ++k) {
        const float d2   = fmaxf(rv[k] + qnorm, 0.0f);
        const float dist = fminf(__fsqrt_rn(d2) * (1.0f / 32.0f), 0.5f);
        const float td   = (0.5f - dist) * 2.0f;
        if (half == 0) { sTmpd[wave][col][k] = td; sIdx[wave][col][k] = ri[k]; }
    }

    // Phase 3: gather + sigmoid-gated sum; wave handles its 16 queries, channels by lane.
    const int ch0 = lane, ch1 = lane + 32, ch2 = lane + 64, ch3 = lane + 96;
    const float w0 = w1[ch0], wA = w1[ch1], wB = w1[ch2];
    const float wC = (ch3 < NCH) ? w1[ch3] : 0.0f;
    const float bb = b1[0];

    for (int qq = 0; qq < 16; ++qq) {
        const int n = n0 + qq;
        const float* fa = feats_a + ((size_t)b * NA + n) * NCH;
        float acc0 = 0.0f, acc1 = 0.0f, acc2 = 0.0f, acc3 = 0.0f;
        #pragma unroll
        for (int k = 0; k < TOPK; ++k) {
            const int   idx = sIdx[wave][qq][k];
            const float td  = sTmpd[wave][qq][k];
            const float* fb = feats_b + ((size_t)b * NB + idx) * NCH;
            const float f0 = fb[ch0], f1 = fb[ch1], f2 = fb[ch2];
            const float f3 = (ch3 < NCH) ? fb[ch3] : 0.0f;
            float p = f0*w0 + f1*wA + f2*wB + f3*wC;
            #pragma unroll
            for (int off = 16; off > 0; off >>= 1) p += __shfl_xor(p, off);
            const float wgt = td / (1.0f + __expf(-(p + bb)));
            acc0 += f0 * wgt; acc1 += f1 * wgt; acc2 += f2 * wgt; acc3 += f3 * wgt;
        }
        float* op = out + ((size_t)b * NA + n) * (2 * NCH);
        op[ch0] = fa[ch0]; op[ch1] = fa[ch1]; op[ch2] = fa[ch2];
        op[NCH + ch0] = acc0; op[NCH + ch1] = acc1; op[NCH + ch2] = acc2;
        if (ch3 < NCH) { op[ch3] = fa[ch3]; op[NCH + ch3] = acc3; }
    }
}

extern "C" void kernel_launch(void* const* d_in, const int* in_sizes, int n_in,
                              void* d_out, int out_size, void* d_ws, size_t ws_size,
                              hipStream_t stream) {
    (void)in_sizes; (void)n_in; (void)d_ws; (void)ws_size; (void)out_size;
    const int*   coords_a = (const int*)  d_in[0];
    const int*   coords_b = (const int*)  d_in[1];
    const float* feats_a  = (const float*)d_in[2];
    const float* feats_b  = (const float*)d_in[3];
    const float* w1       = (const float*)d_in[4];
    const float* b1       = (const float*)d_in[5];
    float*       out      = (float*)d_out;

    const dim3 grid(BATCH * (NA / QPB));   // 4 * 64 = 256 blocks
    const dim3 block(256);                 // 8 waves of 32
    distmatch_topk_wmma<<<grid, block, 0, stream>>>(
        coords_a, coords_b, feats_a, feats_b, w1, b1, out);
}